// AttnAggregator_28518582846056
// MI455X (gfx1250) — compile-verified
//
#include <hip/hip_runtime.h>
#include <hip/hip_bf16.h>
#include <math.h>

#define H 256

typedef __attribute__((ext_vector_type(2))) float v2f;
typedef __attribute__((ext_vector_type(8))) float v8f;

// ---------------------------------------------------------------------------
// C[M,256] = gather(A0,idx0) @ W0  (+ gather(A1,idx1) @ W1)  (+ bias)
// fp32 WMMA 16x16x4. One wave computes a 16x64 strip of C; 8 waves/block
// stack on M; gridDim.y = 4 covers N=256.
// A-frag (16x4 f32): lane = 16*(k>>1)+m, vgpr = k&1  -> contiguous float2/lane
// B-frag (4x16 f32): lane = 16*(k>>1)+n, vgpr = k&1
// C/D    (16x16 f32): vgpr j -> m = j + 8*(lane>>4), n = lane&15
// ---------------------------------------------------------------------------
__global__ __launch_bounds__(256) void wmma_gemm_kernel(
    const float* __restrict__ A0, const int* __restrict__ idx0,
    const float* __restrict__ W0,
    const float* __restrict__ A1, const int* __restrict__ idx1,
    const float* __restrict__ W1,
    const float* __restrict__ bias, float* __restrict__ C, int M)
{
    const int lane = threadIdx.x & 31;
    const int wave = threadIdx.x >> 5;
    const int lm   = lane & 15;   // m (A) / n (B) within tile
    const int lh   = lane >> 4;   // k-half select
    const int m0   = blockIdx.x * 128 + wave * 16;
    const int n0   = blockIdx.y * 64;

    int row  = m0 + lm;
    int crow = row < M ? row : M - 1;   // clamp for tail-tile loads

    v8f acc[4] = {{}, {}, {}, {}};

    {
        const int ar = idx0 ? idx0[crow] : crow;
        const float* __restrict__ Ar = A0 + (size_t)ar * H;
        for (int kb = 0; kb < H; kb += 4) {
            v2f a;
            a.x = Ar[kb + 2 * lh];
            a.y = Ar[kb + 2 * lh + 1];
            const float* __restrict__ Wk = W0 + (size_t)(kb + 2 * lh) * H + n0 + lm;
#pragma unroll
            for (int t = 0; t < 4; ++t) {
                v2f b;
                b.x = Wk[t * 16];
                b.y = Wk[t * 16 + H];
                acc[t] = __builtin_amdgcn_wmma_f32_16x16x4_f32(
                    false, a, false, b, (short)0, acc[t], false, false);
            }
        }
    }
    if (A1) {
        const int ar = idx1 ? idx1[crow] : crow;
        const float* __restrict__ Ar = A1 + (size_t)ar * H;
        for (int kb = 0; kb < H; kb += 4) {
            v2f a;
            a.x = Ar[kb + 2 * lh];
            a.y = Ar[kb + 2 * lh + 1];
            const float* __restrict__ Wk = W1 + (size_t)(kb + 2 * lh) * H + n0 + lm;
#pragma unroll
            for (int t = 0; t < 4; ++t) {
                v2f b;
                b.x = Wk[t * 16];
                b.y = Wk[t * 16 + H];
                acc[t] = __builtin_amdgcn_wmma_f32_16x16x4_f32(
                    false, a, false, b, (short)0, acc[t], false, false);
            }
        }
    }

#pragma unroll
    for (int t = 0; t < 4; ++t) {
        const int n = n0 + t * 16 + lm;
        const float bb = bias ? bias[n] : 0.0f;
#pragma unroll
        for (int j = 0; j < 8; ++j) {
            const int m = m0 + j + 8 * lh;
            if (m < M) C[(size_t)m * H + n] = acc[t][j] + bb;
        }
    }
}

// ---------------------------------------------------------------------------
// scores[n] = sum_h tanh(T[nbr[n]][h] + pre[seg[n]/seq_len][h]) * v_s[h]
// One wave32 per neighbor, 8 h-values per lane, shfl-xor reduction.
// ---------------------------------------------------------------------------
__global__ __launch_bounds__(256) void score_kernel(
    const float* __restrict__ T, const float* __restrict__ pre,
    const int* __restrict__ nbr_ids, const int* __restrict__ seg_ids,
    const float* __restrict__ v_s, float* __restrict__ scores,
    int N, int seq_len)
{
    const int n    = (int)((blockIdx.x * blockDim.x + threadIdx.x) >> 5);
    const int lane = threadIdx.x & 31;
    if (n >= N) return;

    const float* __restrict__ tr = T + (size_t)nbr_ids[n] * H;
    const float* __restrict__ pr = pre + (size_t)(seg_ids[n] / seq_len) * H;

    float acc = 0.0f;
#pragma unroll
    for (int j = 0; j < 8; ++j) {
        const int h = lane + 32 * j;
        acc += tanhf(tr[h] + pr[h]) * v_s[h];
    }
#pragma unroll
    for (int off = 16; off > 0; off >>= 1)
        acc += __shfl_xor(acc, off, 32);
    if (lane == 0) scores[n] = acc;
}

__device__ __forceinline__ int lower_bound_dev(const int* __restrict__ a, int n, int key)
{
    int lo = 0, hi = n;
    while (lo < hi) {
        const int mid = (lo + hi) >> 1;
        if (a[mid] < key) lo = mid + 1; else hi = mid;
    }
    return lo;
}

// ---------------------------------------------------------------------------
// One wave32 per segment: softmax over its neighbors + weighted aggregation
// of em, then write out[seg, 0:768] = [agg | s_emb | r_emb] (zeros if empty).
// ---------------------------------------------------------------------------
__global__ __launch_bounds__(256) void softmax_agg_kernel(
    const float* __restrict__ scores, const int* __restrict__ seg_ids,
    const int* __restrict__ nbr_ids, const float* __restrict__ ent,
    const float* __restrict__ rel, const int* __restrict__ s_idx,
    const int* __restrict__ r_idx, float* __restrict__ out,
    int N, int nseg, int seq_len)
{
    const int seg  = (int)((blockIdx.x * blockDim.x + threadIdx.x) >> 5);
    const int lane = threadIdx.x & 31;
    if (seg >= nseg) return;

    float* __restrict__ orow = out + (size_t)seg * (3 * H);

    const int lo = lower_bound_dev(seg_ids, N, seg);
    const int hi = lower_bound_dev(seg_ids, N, seg + 1);

    if (lo >= hi) {            // empty segment -> fully masked row
#pragma unroll
        for (int j = 0; j < 24; ++j) orow[lane + 32 * j] = 0.0f;
        return;
    }

    // segment max
    float m = -INFINITY;
    for (int i = lo + lane; i < hi; i += 32) m = fmaxf(m, scores[i]);
#pragma unroll
    for (int off = 16; off > 0; off >>= 1)
        m = fmaxf(m, __shfl_xor(m, off, 32));

    // segment exp-sum
    float dsum = 0.0f;
    for (int i = lo + lane; i < hi; i += 32) dsum += expf(scores[i] - m);
#pragma unroll
    for (int off = 16; off > 0; off >>= 1)
        dsum += __shfl_xor(dsum, off, 32);
    const float inv = 1.0f / dsum;

    // weighted aggregation: lanes own h-columns (coalesced em reads)
    float acc[8] = {0.f, 0.f, 0.f, 0.f, 0.f, 0.f, 0.f, 0.f};
    for (int i = lo; i < hi; ++i) {
        const float w = expf(scores[i] - m) * inv;
        const float* __restrict__ er = ent + (size_t)nbr_ids[i] * H;
#pragma unroll
        for (int j = 0; j < 8; ++j) acc[j] += w * er[lane + 32 * j];
    }

    const int b = seg / seq_len;
    const float* __restrict__ se = ent + (size_t)s_idx[b] * H;
    const float* __restrict__ re = rel + (size_t)r_idx[b] * H;
#pragma unroll
    for (int j = 0; j < 8; ++j) {
        const int h = lane + 32 * j;
        orow[h]           = acc[j];
        orow[H + h]       = se[h];
        orow[2 * H + h]   = re[h];
    }
}

extern "C" void kernel_launch(void* const* d_in, const int* in_sizes, int n_in,
                              void* d_out, int out_size, void* d_ws, size_t ws_size,
                              hipStream_t stream)
{
    const int*   s       = (const int*)d_in[0];
    const int*   r       = (const int*)d_in[1];
    const int*   nbr_ids = (const int*)d_in[2];
    const int*   seg_ids = (const int*)d_in[3];
    const float* ent     = (const float*)d_in[4];
    const float* rel     = (const float*)d_in[5];
    const float* W       = (const float*)d_in[6];   // [3H, H] row-major
    const float* b_attn  = (const float*)d_in[7];
    const float* v_s     = (const float*)d_in[8];
    float*       out     = (float*)d_out;

    const int num_ent = in_sizes[4] / H;            // 20000
    const int Bq      = in_sizes[0];                // 2048
    const int N       = in_sizes[2];                // 327680
    const int seq_len = out_size / (Bq * 3 * H);    // 10
    const int nseg    = Bq * seq_len;               // 20480

    // workspace: T[num_ent,H] | pre[Bq,H] | scores[N]   (~24 MB)
    float* T      = (float*)d_ws;
    float* pre    = T + (size_t)num_ent * H;
    float* scores = pre + (size_t)Bq * H;

    const dim3 blk(256, 1, 1);

    // 1) T = ent_embeds @ W_e
    dim3 g1((num_ent + 127) / 128, 4, 1);
    wmma_gemm_kernel<<<g1, blk, 0, stream>>>(
        ent, nullptr, W, nullptr, nullptr, nullptr, nullptr, T, num_ent);

    // 2) pre = ent[s] @ W_s + rel[r] @ W_r + b_attn
    dim3 g2((Bq + 127) / 128, 4, 1);
    wmma_gemm_kernel<<<g2, blk, 0, stream>>>(
        ent, s, W + (size_t)H * H, rel, r, W + (size_t)2 * H * H,
        b_attn, pre, Bq);

    // 3) per-neighbor scores (one wave each)
    const int g3 = (int)(((long long)N * 32 + 255) / 256);
    score_kernel<<<dim3(g3, 1, 1), blk, 0, stream>>>(
        T, pre, nbr_ids, seg_ids, v_s, scores, N, seq_len);

    // 4) segment softmax + aggregation + output assembly (one wave/segment)
    const int g4 = (int)(((long long)nseg * 32 + 255) / 256);
    softmax_agg_kernel<<<dim3(g4, 1, 1), blk, 0, stream>>>(
        scores, seg_ids, nbr_ids, ent, rel, s, r, out, N, nseg, seq_len);
}